// PyGraph_70351564309000
// MI455X (gfx1250) — compile-verified
//
#include <hip/hip_runtime.h>
#include <hip/hip_bf16.h>

// ---------- types ----------
typedef __attribute__((ext_vector_type(2))) float v2f;
typedef __attribute__((ext_vector_type(8))) float v8f;

__device__ __forceinline__ v8f wmma_f32_16x16x4(v2f a, v2f b, v8f c) {
  // D = A(16x4, f32) * B(4x16, f32) + C(16x16, f32)
  return __builtin_amdgcn_wmma_f32_16x16x4_f32(
      /*neg_a=*/false, a, /*neg_b=*/false, b,
      /*c_mod=*/(short)0, c, /*reuse_a=*/false, /*reuse_b=*/false);
}

#define NNODES 8192
#define NC     192
#define KNN    9

struct GInfo { int s[10]; };   // group boundaries s[0..9], s[9] == 8192

// ---------- 1) [B,C,HW] -> node-major x_f[N,C] ----------
__global__ void k_transpose(const float* __restrict__ xin, float* __restrict__ xf) {
  __shared__ float tile[32][33];
  const int b  = blockIdx.z;          // 0..7
  const int ct = blockIdx.y;          // 0..5  (channel tile)
  const int ht = blockIdx.x;          // 0..31 (hw tile)
  const int tx = threadIdx.x;         // 0..31
  const int ty = threadIdx.y;         // 0..7
  const float* src = xin + (size_t)b * NC * 1024;
#pragma unroll
  for (int i = 0; i < 4; ++i) {
    int c  = ct * 32 + ty + i * 8;
    int hw = ht * 32 + tx;
    tile[ty + i * 8][tx] = src[c * 1024 + hw];
  }
  __syncthreads();
#pragma unroll
  for (int i = 0; i < 4; ++i) {
    int hwl = ty + i * 8;
    int n   = b * 1024 + ht * 32 + hwl;
    int c   = ct * 32 + tx;
    xf[(size_t)n * NC + c] = tile[tx][hwl];
  }
}

// ---------- 2) squared norms, one wave per node ----------
__global__ void k_sqnorm(const float* __restrict__ xf, float* __restrict__ sq) {
  const int lane = threadIdx.x & 31;
  const int node = blockIdx.x * 8 + (threadIdx.x >> 5);
  const float* row = xf + (size_t)node * NC;
  float s = 0.f;
#pragma unroll
  for (int u = 0; u < 6; ++u) { float v = row[lane + 32 * u]; s += v * v; }
#pragma unroll
  for (int off = 16; off; off >>= 1) s += __shfl_xor(s, off, 32);
  if (lane == 0) sq[node] = s;
}

// ---------- 3) per-group distance strip (WMMA) + top-9 selection ----------
#define DSTR 1028            // LDS row stride for 16x1024 distance strip
#define ASTR 194             // LDS row stride for 16x192 A strip
__global__ __launch_bounds__(256)
void k_dist_topk(const float* __restrict__ xf, const float* __restrict__ sq,
                 int* __restrict__ srcOut, GInfo gi) {
  __shared__ float s_dist[16 * DSTR];
  __shared__ float s_a[16 * ASTR];

  const int bid   = blockIdx.x;                       // 0..512
  const int g     = (bid >= 512) ? 8 : (bid >> 6);
  const int strip = (bid >= 512) ? 0 : (bid & 63);
  const int gstart = gi.s[g];
  const int gend   = gi.s[g + 1];
  const int cols   = gend - gstart;
  const int rowbase = gstart + strip * 16;
  if (rowbase >= gend) return;                        // uniform per block
  const int ntiles = (cols + 15) >> 4;
  const int scanLimit = ntiles << 4;

  const int tid  = threadIdx.x;
  const int lane = tid & 31;
  const int wv   = tid >> 5;
  const int h    = lane >> 4;                         // lane half selects k pair
  const int lm   = lane & 15;

  // stage A strip (16 query rows x 192) into LDS
  for (int e = tid; e < 16 * NC; e += 256) {
    int r = e / NC, k = e - r * NC;
    int rg = rowbase + r; if (rg >= gend) rg = gend - 1;
    s_a[r * ASTR + k] = xf[(size_t)rg * NC + k];
  }
  __syncthreads();

  const float* aptr = &s_a[lm * ASTR + 2 * h];

  for (int jt = wv; jt < ntiles; jt += 8) {
    const int colbase = gstart + jt * 16;
    int jrow = colbase + lm; if (jrow >= gend) jrow = gend - 1;
    const float* bptr = xf + (size_t)jrow * NC + 2 * h;

    // prefetch the next B tile this wave will consume (global_prefetch_b8)
    {
      int jt2 = jt + 8;
      if (jt2 < ntiles) {
        int jr2 = gstart + jt2 * 16 + lm; if (jr2 >= gend) jr2 = gend - 1;
        __builtin_prefetch(xf + (size_t)jr2 * NC, 0, 0);
      }
    }

    v8f c = {};
#pragma unroll 4
    for (int kk = 0; kk < NC; kk += 4) {
      v2f a = *(const v2f*)(aptr + kk);
      v2f b = *(const v2f*)(bptr + kk);
      c = wmma_f32_16x16x4(a, b, c);
    }
    // d2 = |xi|^2 + |xj|^2 - 2*<xi,xj>
    int cg = colbase + lm; if (cg >= gend) cg = gend - 1;
    const float sqn = sq[cg];
    const int coln = jt * 16 + lm;
#pragma unroll
    for (int vv = 0; vv < 8; ++vv) {
      int m = vv + 8 * h;
      int rg = rowbase + m; if (rg >= gend) rg = gend - 1;
      float dd = sq[rg] + sqn - 2.0f * c[vv];
      if (coln >= cols) dd = __builtin_inff();
      s_dist[m * DSTR + coln] = dd;
    }
  }
  __syncthreads();

  // top-9 smallest per row; wave wv owns rows 2*wv, 2*wv+1
  for (int rr = 0; rr < 2; ++rr) {
    const int r = wv * 2 + rr;
    const int t = rowbase + r;
    const bool rowOK = (t < gend);
    float* drow = &s_dist[r * DSTR];
    for (int it = 0; it < KNN; ++it) {
      float bv = __builtin_inff(); int bi = -1;
      if (rowOK) {
        for (int jc = lane; jc < scanLimit; jc += 32) {
          float v = drow[jc];
          if (v < bv) { bv = v; bi = jc; }
        }
      }
#pragma unroll
      for (int off = 16; off; off >>= 1) {
        float ov = __shfl_xor(bv, off, 32);
        int   oi = __shfl_xor(bi, off, 32);
        if (ov < bv || (ov == bv && (unsigned)oi < (unsigned)bi)) { bv = ov; bi = oi; }
      }
      if (rowOK && lane == 0) {
        srcOut[t * KNN + it] = (bi >= 0 && bv < __builtin_inff()) ? (gstart + bi) : -1;
        if (bi >= 0) drow[bi] = __builtin_inff();   // invalidate for next round
      }
      // LDS is in-order per wave; ensure the invalidating store lands before rescan
      asm volatile("s_wait_dscnt 0" ::: "memory");
    }
  }
}

// ---------- 4) degree ----------
__global__ void k_zero_deg(int* __restrict__ deg) { deg[blockIdx.x * 256 + threadIdx.x] = 0; }

__global__ void k_degree(const int* __restrict__ src, int* __restrict__ deg) {
  const int e = blockIdx.x * 256 + threadIdx.x;      // 8192*9 edges
  const int t = e / KNN;
  const int s = src[e];
  if (s >= 0 && s != t) atomicAdd(&deg[s], 1);
}

__global__ void k_dis(const int* __restrict__ deg, float* __restrict__ dis) {
  const int i = blockIdx.x * 256 + threadIdx.x;
  const int d = deg[i];
  dis[i] = (d > 0) ? __frsqrt_rn((float)d) : 0.0f;
}

// ---------- 5) sparse aggregation: Tx1[t] = sum_j -dis[s]*dis[t] * x_f[s] ----------
__global__ void k_aggregate(const float* __restrict__ xf, const int* __restrict__ src,
                            const float* __restrict__ dis, float* __restrict__ tx1) {
  const int lane = threadIdx.x & 31;
  const int t = blockIdx.x * 8 + (threadIdx.x >> 5);
  const float dt = dis[t];
  float acc[6] = {0.f, 0.f, 0.f, 0.f, 0.f, 0.f};
#pragma unroll
  for (int j = 0; j < KNN; ++j) {
    const int s = src[t * KNN + j];
    if (s >= 0 && s != t) {
      const float coef = -dis[s] * dt;
      const float* row = xf + (size_t)s * NC;
#pragma unroll
      for (int u = 0; u < 6; ++u) acc[u] += coef * row[lane + 32 * u];
    }
  }
  float* orow = tx1 + (size_t)t * NC;
#pragma unroll
  for (int u = 0; u < 6; ++u) orow[lane + 32 * u] = acc[u];
}

// ---------- 6) out = x_f@W0 + Tx1@W1 + bias   (fused WMMA, W staged in LDS) ----------
#define WSTR 194             // LDS stride for transposed 16x192 weight slabs
__global__ __launch_bounds__(256)
void k_outgemm(const float* __restrict__ xf, const float* __restrict__ tx1,
               const float* __restrict__ W0, const float* __restrict__ W1,
               const float* __restrict__ bias, float* __restrict__ out) {
  __shared__ float s_w0[16 * WSTR];   // s_w0[n][k] = W0[k][j0+n]
  __shared__ float s_w1[16 * WSTR];

  const int tid  = threadIdx.x;
  const int lane = tid & 31;
  const int wv   = tid >> 5;
  const int jt   = blockIdx.x % 12;                  // output column tile
  const int seg  = blockIdx.x / 12;                  // 0..7 row segment
  const int j0   = jt * 16;
  const int h    = lane >> 4;
  const int lm   = lane & 15;

  // stage transposed weight slabs: 2 x (16 cols x 192 k)
  for (int e = tid; e < 16 * NC; e += 256) {
    int k = e >> 4, n = e & 15;                      // consecutive tid -> consecutive n
    s_w0[n * WSTR + k] = W0[k * NC + j0 + n];
    s_w1[n * WSTR + k] = W1[k * NC + j0 + n];
  }
  __syncthreads();

  const float* b0p = &s_w0[lm * WSTR + 2 * h];
  const float* b1p = &s_w1[lm * WSTR + 2 * h];
  const float bb = bias[j0 + lm];

  // each wave sweeps 8 row tiles with the staged weights
  for (int q = 0; q < 8; ++q) {
    const int i0 = (seg * 64 + wv * 8 + q) * 16;
    const float* aX = xf  + (size_t)(i0 + lm) * NC + 2 * h;
    const float* aT = tx1 + (size_t)(i0 + lm) * NC + 2 * h;

    v8f c0 = {}, c1 = {};                            // two independent WMMA chains
#pragma unroll 4
    for (int kk = 0; kk < NC; kk += 4) {
      v2f a0 = *(const v2f*)(aX + kk);
      v2f b0 = *(const v2f*)(b0p + kk);
      c0 = wmma_f32_16x16x4(a0, b0, c0);
      v2f a1 = *(const v2f*)(aT + kk);
      v2f b1 = *(const v2f*)(b1p + kk);
      c1 = wmma_f32_16x16x4(a1, b1, c1);
    }
#pragma unroll
    for (int vv = 0; vv < 8; ++vv) {
      const int m = vv + 8 * h;
      out[(size_t)(i0 + m) * NC + j0 + lm] = c0[vv] + c1[vv] + bb;
    }
  }
}

// ---------- launch ----------
extern "C" void kernel_launch(void* const* d_in, const int* in_sizes, int n_in,
                              void* d_out, int out_size, void* d_ws, size_t ws_size,
                              hipStream_t stream) {
  const float* x    = (const float*)d_in[0];   // [8,192,32,32]
  const float* W0   = (const float*)d_in[1];   // [192,192]
  const float* W1   = (const float*)d_in[2];   // [192,192]
  const float* bias = (const float*)d_in[3];   // [192]
  float* out = (float*)d_out;                  // [8192,192]

  char* ws = (char*)d_ws;
  float* xf  = (float*)(ws);                         // 8192*192 f32 = 6291456 B
  float* tx1 = (float*)(ws + 6291456);               // 6291456 B
  float* sq  = (float*)(ws + 12582912);              // 32768 B
  float* dis = (float*)(ws + 12615680);              // 32768 B
  int*   deg = (int*)  (ws + 12648448);              // 32768 B
  int*   src = (int*)  (ws + 12681216);              // 8192*9*4 = 294912 B

  // group boundaries of trunc(linspace(0,8,8192)): start[g] = ceil(g*8191/8)
  GInfo gi;
  for (int g = 0; g < 9; ++g) gi.s[g] = (g * 8191 + 7) / 8;
  gi.s[9] = NNODES;

  k_transpose<<<dim3(32, 6, 8), dim3(32, 8), 0, stream>>>(x, xf);
  k_sqnorm<<<NNODES / 8, 256, 0, stream>>>(xf, sq);
  k_dist_topk<<<513, 256, 0, stream>>>(xf, sq, src, gi);
  k_zero_deg<<<NNODES / 256, 256, 0, stream>>>(deg);
  k_degree<<<NNODES * KNN / 256, 256, 0, stream>>>(src, deg);
  k_dis<<<NNODES / 256, 256, 0, stream>>>(deg, dis);
  k_aggregate<<<NNODES / 8, 256, 0, stream>>>(xf, src, dis, tx1);
  k_outgemm<<<12 * 8, 256, 0, stream>>>(xf, tx1, W0, W1, bias, out);
}